// UpModule_9234179687188
// MI455X (gfx1250) — compile-verified
//
#include <hip/hip_runtime.h>
#include <hip/hip_bf16.h>

typedef __attribute__((ext_vector_type(2))) float v2f;
typedef __attribute__((ext_vector_type(8))) float v8f;

#define Bn 8
#define Cc 128
#define Hh 64
#define Ww 64
#define HW 4096
#define CC 64      // compressed channels
#define EO 100     // encoder output channels (25 taps * 4 subpixels)

#if __has_builtin(__builtin_amdgcn_global_load_async_to_lds_b32)
#define HAVE_ASYNC_LDS 1
#else
#define HAVE_ASYNC_LDS 0
#endif

typedef __attribute__((address_space(1))) int* gptr_i32;
typedef __attribute__((address_space(3))) int* lptr_i32;

__device__ __forceinline__ void wait_asynccnt0() {
#if __has_builtin(__builtin_amdgcn_s_wait_asynccnt)
  __builtin_amdgcn_s_wait_asynccnt(0);
#else
  asm volatile("s_wait_asynccnt 0x0" ::: "memory");
#endif
}

// ---------------------------------------------------------------------------
// Kernel 1: 1x1 conv (channel compressor) as GEMM via V_WMMA_F32_16X16X4_F32.
// M = 64 output channels (4 tiles of 16, one per wave), K = 128 in, N = pixels.
// ---------------------------------------------------------------------------
__global__ __launch_bounds__(128) void k_compress(const float* __restrict__ x,
                                                  const float* __restrict__ w,
                                                  const float* __restrict__ bias,
                                                  float* __restrict__ comp) {
  const int lane = threadIdx.x & 31;
  const int wave = threadIdx.x >> 5;     // 0..3 -> oc tile
  const int pix0 = blockIdx.x * 16;      // 16 consecutive pixels, same batch
  const int b    = pix0 >> 12;
  const int hw0  = pix0 & (HW - 1);
  const int n    = lane & 15;
  const int hi   = lane >> 4;            // lane half selects K pair / M+8 row
  const int m0   = wave * 16;
  const int m    = m0 + n;               // A-matrix row for this lane
  const int koff = hi * 2;

  v8f acc;
#pragma unroll
  for (int r = 0; r < 8; ++r) acc[r] = bias[m0 + r + hi * 8];

  const float* xb = x + (size_t)b * Cc * HW + hw0 + n;
  const float* wr = w + (size_t)m * Cc;

  for (int k0 = 0; k0 < Cc; k0 += 4) {
    v2f a, bb;
    a.x  = wr[k0 + koff];                           // A: 16x4, K=koff
    a.y  = wr[k0 + koff + 1];                       //          K=koff+1
    bb.x = xb[(size_t)(k0 + koff) * HW];            // B: 4x16, row K=koff
    bb.y = xb[(size_t)(k0 + koff + 1) * HW];        //          row K=koff+1
    acc = __builtin_amdgcn_wmma_f32_16x16x4_f32(false, a, false, bb,
                                                (short)0, acc, false, false);
  }

  float* cp = comp + (size_t)b * CC * HW + hw0 + n;
#pragma unroll
  for (int r = 0; r < 8; ++r) {
    const int oc = m0 + r + hi * 8;
    cp[(size_t)oc * HW] = acc[r];
  }
}

// ---------------------------------------------------------------------------
// Kernel 2: 3x3 conv (content encoder) as im2col GEMM via WMMA f32.
// M = 100 (7 wave-tiles of 16, zero-padded to 112), K = 9*64 = 576,
// N = 16 pixels along a row. K order is tap-major so a K-chunk of 4 never
// crosses a (dy,dx) boundary.
// ---------------------------------------------------------------------------
__global__ __launch_bounds__(224) void k_encode(const float* __restrict__ comp,
                                                const float* __restrict__ w,
                                                const float* __restrict__ bias,
                                                float* __restrict__ logits) {
  const int lane = threadIdx.x & 31;
  const int wave = threadIdx.x >> 5;     // 0..6 -> oc tile
  const int tile = blockIdx.x;           // (b, h, w-tile)
  const int wq   = tile & 3;
  const int h    = (tile >> 2) & 63;
  const int b    = tile >> 8;
  const int w0   = wq * 16;
  const int n    = lane & 15;
  const int hi   = lane >> 4;
  const int koff = hi * 2;
  const int m0   = wave * 16;
  const int m    = m0 + n;
  const int wcol = w0 + n;

  v8f acc;
#pragma unroll
  for (int r = 0; r < 8; ++r) {
    const int oc = m0 + r + hi * 8;
    acc[r] = (oc < EO) ? bias[oc] : 0.0f;
  }

  const float* cb = comp + (size_t)b * CC * HW;

  for (int k0 = 0; k0 < 9 * CC; k0 += 4) {
    const int ka = k0 + koff;            // even within a tap block
    const int t  = ka >> 6;              // tap index 0..8 (constant in chunk)
    const int ic = ka & 63;
    const int dy = t / 3 - 1;
    const int dx = t % 3 - 1;

    v2f a, bb;
    if (m < EO) {
      a.x = w[((size_t)(m * CC + ic) * 9) + t];
      a.y = w[((size_t)(m * CC + ic + 1) * 9) + t];
    } else {
      a.x = 0.0f; a.y = 0.0f;
    }

    const int hh = h + dy;
    const int ww = wcol + dx;
    const bool vld = (hh >= 0) & (hh < Hh) & (ww >= 0) & (ww < Ww);
    bb.x = vld ? cb[(size_t)ic * HW + hh * Ww + ww] : 0.0f;
    bb.y = vld ? cb[(size_t)(ic + 1) * HW + hh * Ww + ww] : 0.0f;

    acc = __builtin_amdgcn_wmma_f32_16x16x4_f32(false, a, false, bb,
                                                (short)0, acc, false, false);
  }

  float* lp = logits + (size_t)b * EO * HW + h * Ww + w0 + n;
#pragma unroll
  for (int r = 0; r < 8; ++r) {
    const int oc = m0 + r + hi * 8;
    if (oc < EO) lp[(size_t)oc * HW] = acc[r];
  }
}

// ---------------------------------------------------------------------------
// Kernel 3: softmax over 25 taps (per subpixel) + 5x5 feature reassembly.
// One block per (b, h, 16-wide pixel tile). Logits and the 128ch x 5row x
// 20col x-patch are staged in LDS; the patch is filled with gfx1250 async
// load-to-LDS ops (ASYNCcnt) when available. Inner loop is LDS-only.
// ---------------------------------------------------------------------------
__global__ __launch_bounds__(256) void k_reassemble(const float* __restrict__ x,
                                                    const float* __restrict__ logits,
                                                    float* __restrict__ out) {
  __shared__ float wgt[EO][16];          // [k*4 + (sh*2+sw)][pixel]
  __shared__ float xs[Cc * 5 * 20];      // [c][row p][col], col = w0-2 .. w0+17

  const int tile = blockIdx.x;
  const int wq   = tile & 3;
  const int h    = (tile >> 2) & 63;
  const int b    = tile >> 8;
  const int w0   = wq * 16;
  const int tid  = threadIdx.x;

  // ---- Stage logits for 16 pixels into LDS.
  const float* lp = logits + (size_t)b * EO * HW + h * Ww + w0;
  for (int i = tid; i < EO * 16; i += 256) {
    const int ch = i >> 4, px = i & 15;
    wgt[ch][px] = lp[(size_t)ch * HW + px];
  }

  // ---- Stage the x patch: 128 channels x 5 rows x 20 cols (51.2 KB).
  // Border tiles first zero the halo region; interior tiles skip it.
  const bool border = (h < 2) | (h > Hh - 3) | (w0 == 0) | (w0 == Ww - 16);
  if (border) {
    for (int i = tid; i < Cc * 5 * 20; i += 256) xs[i] = 0.0f;
    __syncthreads();
  }
  const float* xbase = x + (size_t)b * Cc * HW;
  for (int i = tid; i < Cc * 5 * 20; i += 256) {
    const int c   = i / 100;
    const int r   = i - c * 100;
    const int p   = r / 20;
    const int col = r - p * 20;
    const int hh  = h + p - 2;
    const int ww  = w0 - 2 + col;
    const bool v  = (hh >= 0) & (hh < Hh) & (ww >= 0) & (ww < Ww);
#if HAVE_ASYNC_LDS
    if (v) {
      const float* gp = xbase + (size_t)c * HW + hh * Ww + ww;
      __builtin_amdgcn_global_load_async_to_lds_b32(
          (gptr_i32)gp, (lptr_i32)&xs[i], 0, 0);
    }
#else
    if (v | !border) xs[i] = v ? xbase[(size_t)c * HW + hh * Ww + ww] : 0.0f;
#endif
  }
#if HAVE_ASYNC_LDS
  wait_asynccnt0();                      // my async loads landed in LDS
#endif
  __syncthreads();                       // everyone's loads visible

  // ---- Softmax over the 25 taps for each (pixel, subpixel): 64 items.
  if (tid < 64) {
    const int px = tid >> 2, sub = tid & 3;
    float mx = -1e30f;
#pragma unroll
    for (int k = 0; k < 25; ++k) mx = fmaxf(mx, wgt[k * 4 + sub][px]);
    float e[25];
    float s = 0.0f;
#pragma unroll
    for (int k = 0; k < 25; ++k) {
      e[k] = __expf(wgt[k * 4 + sub][px] - mx);
      s += e[k];
    }
    const float inv = 1.0f / s;
#pragma unroll
    for (int k = 0; k < 25; ++k) wgt[k * 4 + sub][px] = e[k] * inv;
  }
  __syncthreads();

  // ---- Reassembly: 128 channels x 16 pixels; pure LDS reads + FMAs.
  for (int it = tid; it < Cc * 16; it += 256) {
    const int c  = it >> 4;
    const int px = it & 15;
    const float* xc = &xs[c * 100];

    float a0 = 0.f, a1 = 0.f, a2 = 0.f, a3 = 0.f;
#pragma unroll
    for (int p = 0; p < 5; ++p) {
#pragma unroll
      for (int q = 0; q < 5; ++q) {
        const float xv = xc[p * 20 + px + q];
        const int k4 = (p * 5 + q) * 4;
        a0 += xv * wgt[k4 + 0][px];
        a1 += xv * wgt[k4 + 1][px];
        a2 += xv * wgt[k4 + 2][px];
        a3 += xv * wgt[k4 + 3][px];
      }
    }
    float* op = out + (size_t)(b * Cc + c) * (2 * Hh) * (2 * Ww)
                    + (size_t)(2 * h) * (2 * Ww) + 2 * (w0 + px);
    op[0] = a0;
    op[1] = a1;
    op[2 * Ww] = a2;
    op[2 * Ww + 1] = a3;
  }
}

// ---------------------------------------------------------------------------
extern "C" void kernel_launch(void* const* d_in, const int* in_sizes, int n_in,
                              void* d_out, int out_size, void* d_ws, size_t ws_size,
                              hipStream_t stream) {
  const float* x      = (const float*)d_in[0];
  const float* w_comp = (const float*)d_in[1];
  const float* b_comp = (const float*)d_in[2];
  const float* w_enc  = (const float*)d_in[3];
  const float* b_enc  = (const float*)d_in[4];
  float* out = (float*)d_out;

  float* comp   = (float*)d_ws;                       // 8*64*4096 floats (8 MB)
  float* logits = comp + (size_t)Bn * CC * HW;        // 8*100*4096 floats (13 MB)

  // 2048 pixel tiles of 16 in each kernel's N dimension.
  k_compress<<<2048, 128, 0, stream>>>(x, w_comp, b_comp, comp);
  k_encode<<<2048, 224, 0, stream>>>(comp, w_enc, b_enc, logits);
  k_reassemble<<<2048, 256, 0, stream>>>(x, logits, out);
}